// SpinSphericalBlock_31310311588284
// MI455X (gfx1250) — compile-verified
//
#include <hip/hip_runtime.h>
#include <hip/hip_bf16.h>

// ---------------------------------------------------------------------------
// SpinSphericalBlock on gfx1250 (MI455X), fp32 WMMA + async LDS staging.
//   B=8, RES=64, RES_OUT=32, L_OUT=15, K_OUT=256, SPINS_IN=2, SPINS_OUT=3,
//   C_IN=64, C_OUT=128.
// Pipeline:
//   [0] gen_wigner   : Wigner small-d tables (in & out grids)
//   [1] gen_yin      : Y_IN_W conj*quad  -> [i][tp=4096][k=256] complex planes
//   [2] gen_yout     : Y_OUT             -> [s][tp=1024][k=256] complex planes
//   [3] cgemm<true>  : analysis  C[u,k] = sum_tp x^T * Yin   (WMMA f32, Ksplit 4)
//   [4] mix          : out_coeffs[b,k,s,d], sums K-split partials  (VALU)
//   [5] cgemm<false> : synthesis C[tp,d] = Yout * out_coeffs  (WMMA f32)
//   [6] reduce_stats : per (s,d) sums over (b,tp)
//   [7] finalize     : complex BN + magnitude ReLU -> d_out (interleaved c64)
// ---------------------------------------------------------------------------

typedef __attribute__((ext_vector_type(2))) float v2f;
typedef __attribute__((ext_vector_type(8))) float v8f;

#define PI_F 3.14159265358979323846f
#define KSPLIT1 4

#ifndef __has_builtin
#define __has_builtin(x) 0
#endif
#if __has_builtin(__builtin_amdgcn_global_load_async_to_lds_b128) && \
    __has_builtin(__builtin_amdgcn_s_wait_asynccnt)
#define HAVE_ASYNC_LDS 1
#else
#define HAVE_ASYNC_LDS 0
#endif

// ---------------- workspace layout (float offsets) ----------------
#define N_DIN    (2L*256*64)
#define N_DOUT   (3L*256*32)
#define N_YIN    (2L*4096*256)           // per complex plane
#define N_YOUT   (3L*1024*256)
#define N_COEF   (8L*2*64*256)           // [b][i][u][k]
#define N_COEFP  ((long)KSPLIT1 * N_COEF) // K-split partials
#define N_OC     (8L*256*3*128)          // [b][k][s][d]
#define N_FM     (8L*1024*3*128)         // [b][tp][s][d]

#define OFF_DIN   0L
#define OFF_DOUT  (OFF_DIN   + N_DIN)
#define OFF_YINR  (OFF_DOUT  + N_DOUT)
#define OFF_YINI  (OFF_YINR  + N_YIN)
#define OFF_YOUTR (OFF_YINI  + N_YIN)
#define OFF_YOUTI (OFF_YOUTR + N_YOUT)
#define OFF_CR    (OFF_YOUTI + N_YOUT)
#define OFF_CI    (OFF_CR    + N_COEFP)
#define OFF_OCR   (OFF_CI    + N_COEFP)
#define OFF_OCI   (OFF_OCR   + N_OC)
#define OFF_FMR   (OFF_OCI   + N_OC)
#define OFF_FMI   (OFF_FMR   + N_FM)
#define OFF_STATS (OFF_FMI   + N_FM)

__device__ __forceinline__ int isqrt_small(int k) {
    int l = (int)sqrtf((float)k + 0.5f);
    while (l * l > k) --l;
    while ((l + 1) * (l + 1) <= k) ++l;
    return l;
}

// 16-byte global -> LDS transfer: async on gfx1250 (ASYNCcnt), sync fallback.
// Builtin signature (per hipcc diagnostic): (v4i AS1*, v4i AS3*, imm, imm).
__device__ __forceinline__ void copy16(const float* g, float* l) {
#if HAVE_ASYNC_LDS
    typedef int v4i_t __attribute__((vector_size(16)));
    typedef __attribute__((address_space(1))) v4i_t* gp_t;
    typedef __attribute__((address_space(3))) v4i_t* lp_t;
    gp_t gp = (gp_t)(unsigned long long)(uintptr_t)g;
    lp_t lp = (lp_t)(unsigned int)(uintptr_t)l;  // low 32 bits = LDS offset
    __builtin_amdgcn_global_load_async_to_lds_b128(gp, lp, 0, 0);
#else
    *(float4*)l = *(const float4*)g;
#endif
}

// --------------------------------------------------------------------------
// [0] Wigner small-d column d^l_{m,-s}(theta_t), k = l^2 + (m+l), l <= 15.
// --------------------------------------------------------------------------
__global__ void gen_wigner(float* __restrict__ dTab, int nSpin, int res) {
    long flat = (long)blockIdx.x * blockDim.x + threadIdx.x;
    long total = (long)nSpin * 256 * res;
    if (flat >= total) return;
    int t = (int)(flat % res);
    int k = (int)((flat / res) % 256);
    int s = (int)(flat / ((long)res * 256));

    int l = isqrt_small(k);
    int m = k - l * l - l;
    int n = -s;
    float val = 0.0f;
    if (l >= s) {
        float theta = ((float)t + 0.5f) * PI_F / (float)res;
        float cb = cosf(0.5f * theta);
        float sb = sinf(0.5f * theta);
        float pref = 0.5f * (lgammaf((float)(l + m + 1)) + lgammaf((float)(l - m + 1)) +
                             lgammaf((float)(l + n + 1)) + lgammaf((float)(l - n + 1)));
        int klo = max(0, n - m);
        int khi = min(l + n, l - m);
        for (int kk = klo; kk <= khi; ++kk) {
            float lg = pref - (lgammaf((float)(l + n - kk + 1)) + lgammaf((float)(kk + 1)) +
                               lgammaf((float)(m - n + kk + 1)) + lgammaf((float)(l - m - kk + 1)));
            float sgn = ((m - n + kk) & 1) ? -1.0f : 1.0f;
            val += sgn * expf(lg) * powf(cb, (float)(2 * l + n - m - 2 * kk))
                                  * powf(sb, (float)(m - n + 2 * kk));
        }
    }
    dTab[((long)s * 256 + k) * res + t] = val;
}

// --------------------------------------------------------------------------
// [1] Y_IN_W = conj(Y) * quad_w, layout [i][tp][k].
// --------------------------------------------------------------------------
__global__ void gen_yin(const float* __restrict__ dIn,
                        float* __restrict__ Yr, float* __restrict__ Yi) {
    long flat = (long)blockIdx.x * blockDim.x + threadIdx.x;
    if (flat >= 2L * 4096 * 256) return;
    int k  = (int)(flat % 256);
    int tp = (int)((flat / 256) % 4096);
    int i  = (int)(flat / (256L * 4096));
    int t = tp >> 6, p = tp & 63;
    int l = isqrt_small(k);
    int m = k - l * l - l;
    float d = dIn[((long)i * 256 + k) * 64 + t];
    float theta = ((float)t + 0.5f) * PI_F / 64.0f;
    float w = sinf(theta) * (PI_F / 64.0f) * (2.0f * PI_F / 64.0f);
    float norm = sqrtf((float)(2 * l + 1) / (4.0f * PI_F));
    float amp = ((m & 1) ? -1.0f : 1.0f) * norm * d * w;
    float ang = -(float)m * (float)p * (2.0f * PI_F / 64.0f);  // conj phase
    float sv, cv;
    sincosf(ang, &sv, &cv);
    long o = ((long)i * 4096 + tp) * 256 + k;
    Yr[o] = amp * cv;
    Yi[o] = amp * sv;
}

// --------------------------------------------------------------------------
// [2] Y_OUT, layout [s][tp][k].
// --------------------------------------------------------------------------
__global__ void gen_yout(const float* __restrict__ dOut,
                         float* __restrict__ Yr, float* __restrict__ Yi) {
    long flat = (long)blockIdx.x * blockDim.x + threadIdx.x;
    if (flat >= 3L * 1024 * 256) return;
    int k  = (int)(flat % 256);
    int tp = (int)((flat / 256) % 1024);
    int s  = (int)(flat / (256L * 1024));
    int t = tp >> 5, p = tp & 31;
    int l = isqrt_small(k);
    int m = k - l * l - l;
    float d = dOut[((long)s * 256 + k) * 32 + t];
    float norm = sqrtf((float)(2 * l + 1) / (4.0f * PI_F));
    float amp = ((m & 1) ? -1.0f : 1.0f) * norm * d;
    float ang = (float)m * (float)p * (2.0f * PI_F / 32.0f);
    float sv, cv;
    sincosf(ang, &sv, &cv);
    long o = ((long)s * 1024 + tp) * 256 + k;
    Yr[o] = amp * cv;
    Yi[o] = amp * sv;
}

// --------------------------------------------------------------------------
// [3]/[5] Complex GEMM via V_WMMA_F32_16X16X4_F32, double-buffered async LDS.
//   ATRANS=true : A stored K-major [K][M] (analysis / x); LDS A = [kk][68pad]
//   ATRANS=false: A stored M-major [M][K] (synthesis / Yout); LDS A = [m][20pad]
//   B always [K][N] -> LDS [kk][68pad].  Block tile 64x64, KB=16, 8 waves
//   (4Mx2N).  Per wave per tile: 4 async B128 instructions => prefetch next
//   tile then s_wait_asynccnt(4) retires current (async loads done in order).
// --------------------------------------------------------------------------
template <bool ATRANS>
__global__ __launch_bounds__(256) void cgemm_wmma(
    const float* __restrict__ Ar, const float* __restrict__ Ai,
    const float* __restrict__ Br, const float* __restrict__ Bi,
    float* __restrict__ Cr, float* __restrict__ Ci,
    int K, int lda, int ldb, int ldc,
    long aSB, long aSS, long bSB, long bSS, long cSB, long cSS,
    int S, int ksplit, long cPS) {
    // ATRANS : A 2x[16][68] planes @0,2176 ; B 2x[16][68] planes @4352,6528
    // !ATRANS: A 2x[64][20] planes @0,2560 ; B 2x[16][68] planes @5120,7296
    __shared__ __align__(16) float smem[9472];

    const int tid  = threadIdx.x;
    const int lane = tid & 31;
    const int wid  = tid >> 5;
    const int wm = (wid & 3) * 16;
    const int wn = (wid >> 2) * 32;
    const int n0 = blockIdx.x * 64;
    const int m0 = blockIdx.y * 64;
    int z = blockIdx.z;
    const int split = z % ksplit;
    z /= ksplit;
    const int bb = z / S, ss = z % S;
    const int Kc = K / ksplit;
    const int kBeg = split * Kc, kEnd = kBeg + Kc;

    const float* aR = Ar + bb * aSB + ss * aSS;
    const float* aI = Ai + bb * aSB + ss * aSS;
    const float* bR = Br + bb * bSB + ss * bSS;
    const float* bI = Bi + bb * bSB + ss * bSS;

    float* sAr = smem;
    float* sAi = smem + (ATRANS ? 2176 : 2560);
    float* sBr = smem + (ATRANS ? 4352 : 5120);
    float* sBi = smem + (ATRANS ? 6528 : 7296);
    const int aBufStride = ATRANS ? 1088 : 1280;  // floats per buffer

    auto stage = [&](int buf, int k0) {
        if (ATRANS) {
            int kk = tid >> 4, c4 = (tid & 15) << 2;
            long g = (long)(k0 + kk) * lda + m0 + c4;
            int lo = buf * 1088 + kk * 68 + c4;
            copy16(aR + g, sAr + lo);
            copy16(aI + g, sAi + lo);
        } else {
            int m = tid >> 2, k4 = (tid & 3) << 2;
            long g = (long)(m0 + m) * lda + k0 + k4;
            int lo = buf * 1280 + m * 20 + k4;
            copy16(aR + g, sAr + lo);
            copy16(aI + g, sAi + lo);
        }
        int kk = tid >> 4, n4 = (tid & 15) << 2;
        long g = (long)(k0 + kk) * ldb + n0 + n4;
        int lo = buf * 1088 + kk * 68 + n4;
        copy16(bR + g, sBr + lo);
        copy16(bI + g, sBi + lo);
    };

    v8f cr0 = {}, ci0 = {}, cr1 = {}, ci1 = {};
    const int ml = lane & 15;
    const int kh = (lane >> 4) << 1;  // 0 | 2

    stage(0, kBeg);
    int buf = 0;
    for (int k0 = kBeg; k0 < kEnd; k0 += 16) {
        if (k0 + 16 < kEnd) {
            stage(buf ^ 1, k0 + 16);
#if HAVE_ASYNC_LDS
            __builtin_amdgcn_s_wait_asynccnt(4);  // current tile's 4 ops done
#endif
        } else {
#if HAVE_ASYNC_LDS
            __builtin_amdgcn_s_wait_asynccnt(0);
#endif
        }
        __syncthreads();

        const float* Abr = sAr + buf * aBufStride;
        const float* Abi = sAi + buf * aBufStride;
        const float* Bbr = sBr + buf * 1088;
        const float* Bbi = sBi + buf * 1088;
#pragma unroll
        for (int ks = 0; ks < 16; ks += 4) {
            int kr0 = ks + kh, kr1 = kr0 + 1;
            v2f ar, ai;
            if (ATRANS) {
                ar = v2f{Abr[kr0 * 68 + wm + ml], Abr[kr1 * 68 + wm + ml]};
                ai = v2f{Abi[kr0 * 68 + wm + ml], Abi[kr1 * 68 + wm + ml]};
            } else {
                ar = v2f{Abr[(wm + ml) * 20 + kr0], Abr[(wm + ml) * 20 + kr1]};
                ai = v2f{Abi[(wm + ml) * 20 + kr0], Abi[(wm + ml) * 20 + kr1]};
            }
            v2f an = -ai;  // VALU negate (f32 WMMA NEG bits are C-neg only)
            v2f br0 = {Bbr[kr0 * 68 + wn + ml], Bbr[kr1 * 68 + wn + ml]};
            v2f bi0 = {Bbi[kr0 * 68 + wn + ml], Bbi[kr1 * 68 + wn + ml]};
            v2f br1 = {Bbr[kr0 * 68 + wn + 16 + ml], Bbr[kr1 * 68 + wn + 16 + ml]};
            v2f bi1 = {Bbi[kr0 * 68 + wn + 16 + ml], Bbi[kr1 * 68 + wn + 16 + ml]};
            cr0 = __builtin_amdgcn_wmma_f32_16x16x4_f32(false, ar, false, br0, 0, cr0, false, false);
            cr0 = __builtin_amdgcn_wmma_f32_16x16x4_f32(false, an, false, bi0, 0, cr0, false, false);
            ci0 = __builtin_amdgcn_wmma_f32_16x16x4_f32(false, ar, false, bi0, 0, ci0, false, false);
            ci0 = __builtin_amdgcn_wmma_f32_16x16x4_f32(false, ai, false, br0, 0, ci0, false, false);
            cr1 = __builtin_amdgcn_wmma_f32_16x16x4_f32(false, ar, false, br1, 0, cr1, false, false);
            cr1 = __builtin_amdgcn_wmma_f32_16x16x4_f32(false, an, false, bi1, 0, cr1, false, false);
            ci1 = __builtin_amdgcn_wmma_f32_16x16x4_f32(false, ar, false, bi1, 0, ci1, false, false);
            ci1 = __builtin_amdgcn_wmma_f32_16x16x4_f32(false, ai, false, br1, 0, ci1, false, false);
        }
        __syncthreads();
        buf ^= 1;
    }

    float* cRp = Cr + bb * cSB + ss * cSS + split * cPS;
    float* cIp = Ci + bb * cSB + ss * cSS + split * cPS;
    const int row = m0 + wm + (lane >> 4) * 8;
    const int col = n0 + wn + (lane & 15);
#pragma unroll
    for (int j = 0; j < 8; ++j) {
        long g0 = (long)(row + j) * ldc + col;
        cRp[g0] = cr0[j];
        cIp[g0] = ci0[j];
        cRp[g0 + 16] = cr1[j];
        cIp[g0 + 16] = ci1[j];
    }
}

// --------------------------------------------------------------------------
// [4] Mixing: out_coeffs[b,k,s,d] = sum_{i,u} coeffs[b,k,i,u]*kern[l(k),i,s,u,d]
// Sums the KSPLIT1 analysis partials while loading coeff rows into LDS.
// --------------------------------------------------------------------------
__global__ __launch_bounds__(384) void mix_kernel(
    const float* __restrict__ cR, const float* __restrict__ cI,
    const float* __restrict__ kR, const float* __restrict__ kI,
    float* __restrict__ oR, float* __restrict__ oI) {
    __shared__ float scr[8][128], sci[8][128];
    const int b = blockIdx.z;
    const int l = blockIdx.y;
    const int r0 = blockIdx.x * 8;
    const int nm = 2 * l + 1;
    if (r0 >= nm) return;
    const int tid = threadIdx.x;

    for (int idx = tid; idx < 8 * 128; idx += 384) {
        int r = idx >> 7, iu = idx & 127;
        int i = iu >> 6, u = iu & 63;
        bool valid = (r0 + r) < nm;
        int k = l * l + min(r0 + r, nm - 1);
        long g = (((long)(b * 2 + i)) * 64 + u) * 256 + k;
        float vr = 0.0f, vi = 0.0f;
#pragma unroll
        for (int p2 = 0; p2 < KSPLIT1; ++p2) {
            vr += cR[(long)p2 * N_COEF + g];
            vi += cI[(long)p2 * N_COEF + g];
        }
        scr[r][iu] = valid ? vr : 0.0f;
        sci[r][iu] = valid ? vi : 0.0f;
    }
    __syncthreads();

    const int s = tid >> 7;
    const int d = tid & 127;
    float ar[8] = {}, aii[8] = {};
    for (int iu = 0; iu < 128; ++iu) {
        int i = iu >> 6, u = iu & 63;
        long kg = ((((long)l * 2 + i) * 3 + s) * 64 + u) * 128 + d;
        float kr = kR[kg], ki = kI[kg];
#pragma unroll
        for (int r = 0; r < 8; ++r) {
            ar[r]  = fmaf(scr[r][iu], kr, ar[r]);
            ar[r]  = fmaf(-sci[r][iu], ki, ar[r]);
            aii[r] = fmaf(scr[r][iu], ki, aii[r]);
            aii[r] = fmaf(sci[r][iu], kr, aii[r]);
        }
    }
#pragma unroll
    for (int r = 0; r < 8; ++r) {
        if (r0 + r < nm) {
            int k = l * l + r0 + r;
            long g = (((long)b * 256 + k) * 3 + s) * 128 + d;
            oR[g] = ar[r];
            oI[g] = aii[r];
        }
    }
}

// --------------------------------------------------------------------------
// [6] Per-(s,d) sums over (b,tp): sum_re, sum_im, sum_|.|^2.
// --------------------------------------------------------------------------
__global__ __launch_bounds__(256) void reduce_stats(
    const float* __restrict__ fmR, const float* __restrict__ fmI,
    float* __restrict__ stats) {
    const int col = blockIdx.x;
    const int tid = threadIdx.x;
    float sr = 0.0f, si = 0.0f, sq = 0.0f;
    for (int r = tid; r < 8192; r += 256) {
        long idx = (long)r * 384 + col;
        float re = fmR[idx], im = fmI[idx];
        sr += re;
        si += im;
        sq += re * re + im * im;
    }
    __shared__ float red0[256], red1[256], red2[256];
    red0[tid] = sr;
    red1[tid] = si;
    red2[tid] = sq;
    __syncthreads();
    for (int off = 128; off > 0; off >>= 1) {
        if (tid < off) {
            red0[tid] += red0[tid + off];
            red1[tid] += red1[tid + off];
            red2[tid] += red2[tid + off];
        }
        __syncthreads();
    }
    if (tid == 0) {
        stats[col * 3 + 0] = red0[0];
        stats[col * 3 + 1] = red1[0];
        stats[col * 3 + 2] = red2[0];
    }
}

// --------------------------------------------------------------------------
// [7] Complex BN (mean only for spin 0) + magnitude ReLU; interleaved c64 out.
// --------------------------------------------------------------------------
__global__ __launch_bounds__(256) void finalize(
    const float* __restrict__ fmR, const float* __restrict__ fmI,
    const float* __restrict__ stats, const float* __restrict__ gamma,
    const float* __restrict__ betaR, const float* __restrict__ betaI,
    const float* __restrict__ bias, float* __restrict__ out) {
    long flat = (long)blockIdx.x * blockDim.x + threadIdx.x;
    if (flat >= N_FM) return;
    int col = (int)(flat % 384);
    int s = col >> 7;
    float re = fmR[flat], im = fmI[flat];
    const float invN = 1.0f / 8192.0f;
    float mr = 0.0f, mi = 0.0f;
    if (s == 0) {
        mr = stats[col * 3 + 0] * invN;
        mi = stats[col * 3 + 1] * invN;
    }
    float var = stats[col * 3 + 2] * invN - mr * mr - mi * mi;
    float g = gamma[col] * rsqrtf(var + 1e-5f);
    float yr = (re - mr) * g + (s == 0 ? betaR[col] : 0.0f);
    float yi = (im - mi) * g + (s == 0 ? betaI[col] : 0.0f);
    float mag = sqrtf(yr * yr + yi * yi);
    float f = fmaxf(mag + bias[col], 0.0f) / (mag + 1e-6f);
    out[2 * flat + 0] = yr * f;
    out[2 * flat + 1] = yi * f;
}

// --------------------------------------------------------------------------
extern "C" void kernel_launch(void* const* d_in, const int* in_sizes, int n_in,
                              void* d_out, int out_size, void* d_ws, size_t ws_size,
                              hipStream_t stream) {
    const float* x_real = (const float*)d_in[0];
    const float* x_imag = (const float*)d_in[1];
    const float* kern_r = (const float*)d_in[2];
    const float* kern_i = (const float*)d_in[3];
    const float* gamma  = (const float*)d_in[4];
    const float* beta_r = (const float*)d_in[5];
    const float* beta_i = (const float*)d_in[6];
    const float* nlbias = (const float*)d_in[7];
    float* out = (float*)d_out;

    float* W = (float*)d_ws;
    float* dIn   = W + OFF_DIN;
    float* dOut  = W + OFF_DOUT;
    float* YinR  = W + OFF_YINR;
    float* YinI  = W + OFF_YINI;
    float* YoutR = W + OFF_YOUTR;
    float* YoutI = W + OFF_YOUTI;
    float* cR    = W + OFF_CR;
    float* cI    = W + OFF_CI;
    float* ocR   = W + OFF_OCR;
    float* ocI   = W + OFF_OCI;
    float* fmR   = W + OFF_FMR;
    float* fmI   = W + OFF_FMI;
    float* stats = W + OFF_STATS;

    gen_wigner<<<(int)((N_DIN + 255) / 256), 256, 0, stream>>>(dIn, 2, 64);
    gen_wigner<<<(int)((N_DOUT + 255) / 256), 256, 0, stream>>>(dOut, 3, 32);
    gen_yin<<<(int)((N_YIN + 255) / 256), 256, 0, stream>>>(dIn, YinR, YinI);
    gen_yout<<<(int)((N_YOUT + 255) / 256), 256, 0, stream>>>(dOut, YoutR, YoutI);

    // [3] Analysis: per (b,i,split): C[u=64,k=256] = sum_tp x^T * Yin
    cgemm_wmma<true><<<dim3(256 / 64, 64 / 64, 16 * KSPLIT1), 256, 0, stream>>>(
        x_real, x_imag, YinR, YinI, cR, cI,
        /*K=*/4096, /*lda=*/128, /*ldb=*/256, /*ldc=*/256,
        /*aSB=*/524288L, /*aSS=*/64L, /*bSB=*/0L, /*bSS=*/1048576L,
        /*cSB=*/32768L, /*cSS=*/16384L, /*S=*/2, /*ksplit=*/KSPLIT1, /*cPS=*/N_COEF);

    // [4] Channel mixing (sums K-split partials)
    mix_kernel<<<dim3(4, 16, 8), 384, 0, stream>>>(cR, cI, kern_r, kern_i, ocR, ocI);

    // [5] Synthesis: per (b,s): C[tp=1024,d=128] = Yout * oc
    cgemm_wmma<false><<<dim3(128 / 64, 1024 / 64, 24), 256, 0, stream>>>(
        YoutR, YoutI, ocR, ocI, fmR, fmI,
        /*K=*/256, /*lda=*/256, /*ldb=*/384, /*ldc=*/384,
        /*aSB=*/0L, /*aSS=*/262144L, /*bSB=*/98304L, /*bSS=*/128L,
        /*cSB=*/393216L, /*cSS=*/128L, /*S=*/3, /*ksplit=*/1, /*cPS=*/0L);

    reduce_stats<<<384, 256, 0, stream>>>(fmR, fmI, stats);
    finalize<<<(int)((N_FM + 255) / 256), 256, 0, stream>>>(
        fmR, fmI, stats, gamma, beta_r, beta_i, nlbias, out);
}